// MLA_52828097740984
// MI455X (gfx1250) — compile-verified
//
#include <hip/hip_runtime.h>
#include <hip/hip_bf16.h>

typedef __attribute__((ext_vector_type(2))) float v2f;
typedef __attribute__((ext_vector_type(8))) float v8f;

#define SCALE_F 0.07216878364870323f  // 1/sqrt(192)

__device__ __forceinline__ v8f v8f_zero() {
  v8f z = {0.f, 0.f, 0.f, 0.f, 0.f, 0.f, 0.f, 0.f};
  return z;
}

// D = A(16x4) * B(4x16) + C, fp32 WMMA.
// A VGPR0 = K{0|2} lanes {0-15|16-31}, VGPR1 = K{1|3}; M = lane%16.
// B VGPR0 = K{0|2}, VGPR1 = K{1|3}; N = lane%16.
// C/D VGPR j = row {j | j+8}, N = lane%16.
__device__ __forceinline__ v8f wmma_f32(v2f a, v2f b, v8f c) {
  return __builtin_amdgcn_wmma_f32_16x16x4_f32(false, a, false, b, (short)0, c,
                                               false, false);
}

// ---- CDNA5 async global->LDS copies (ASYNCcnt-tracked, no VGPR staging) ----
__device__ __forceinline__ uint32_t lds_off(const void* p) {
  return (uint32_t)(uintptr_t)p;  // LDS aperture: addr[31:0] == LDS byte offset
}
__device__ __forceinline__ void async_ld_b128(uint32_t lds_addr, const void* ga) {
  asm volatile("global_load_async_to_lds_b128 %0, %1, off"
               :: "v"(lds_addr), "v"(ga) : "memory");
}
__device__ __forceinline__ void async_ld_b64(uint32_t lds_addr, const void* ga) {
  asm volatile("global_load_async_to_lds_b64 %0, %1, off"
               :: "v"(lds_addr), "v"(ga) : "memory");
}
__device__ __forceinline__ void wait_async0() {
  asm volatile("s_wait_asynccnt 0x0" ::: "memory");
}

// ---------------------------------------------------------------------------
// C[M x N] = X[M x K] * W[N x K]^T   (M = 2048 rows total)
// grid: (N/128, M/256), block 256 = 8 waves; wave owns 32 M-rows x 128 N
// (two M sub-blocks -> each B fragment feeds 2 independent WMMA chains).
// Double-buffered LDS, async-to-LDS prefetch pipelined against WMMA compute.
// ---------------------------------------------------------------------------
__global__ __launch_bounds__(256) void mla_gemm_nt(const float* __restrict__ X,
                                                   const float* __restrict__ W,
                                                   float* __restrict__ C,
                                                   int K, int N) {
  __shared__ float lA[2][256 * 20];  // [m][k] stride 20 (bank-conflict free)
  __shared__ float lB[2][128 * 20];  // [n][k] stride 20
  const int tid = threadIdx.x;
  const int wave = tid >> 5;
  const int lane = tid & 31;
  const int hlf = lane >> 4;
  const int l16 = lane & 15;
  const int n0 = blockIdx.x * 128;
  const int m0 = blockIdx.y * 256;

  // Async staging roles: wave w stages A rows [w*32, w*32+32) as 4 b128 ops
  // and B rows [w*16, w*16+16) as 2 b128 ops (8 rows x 16 floats per op).
  const int arow = wave * 32 + (lane >> 2);
  const int brow = wave * 16 + (lane >> 2);
  const int scol = (lane & 3) * 4;
  const float* gA[4];
  const float* gB[2];
  uint32_t lAo[4], lBo[2];
#pragma unroll
  for (int s = 0; s < 4; ++s) {
    gA[s] = X + (size_t)(m0 + arow + s * 8) * K + scol;
    lAo[s] = lds_off(&lA[0][(arow + s * 8) * 20 + scol]);
  }
#pragma unroll
  for (int s = 0; s < 2; ++s) {
    gB[s] = W + (size_t)(n0 + brow + s * 8) * K + scol;
    lBo[s] = lds_off(&lB[0][(brow + s * 8) * 20 + scol]);
  }
  const uint32_t ABUF = 256 * 20 * 4;
  const uint32_t BBUF = 128 * 20 * 4;

  v8f acc[2][8];
#pragma unroll
  for (int ms = 0; ms < 2; ++ms)
#pragma unroll
    for (int t = 0; t < 8; ++t) acc[ms][t] = v8f_zero();

  auto issue = [&](int dstbuf) {
#pragma unroll
    for (int s = 0; s < 4; ++s) {
      async_ld_b128(lAo[s] + (uint32_t)dstbuf * ABUF, gA[s]);
      gA[s] += 16;
    }
#pragma unroll
    for (int s = 0; s < 2; ++s) {
      async_ld_b128(lBo[s] + (uint32_t)dstbuf * BBUF, gB[s]);
      gB[s] += 16;
    }
  };

  // prologue: tile 0 -> buffer 0
  issue(0);
  wait_async0();
  __syncthreads();

  const int ntiles = K >> 4;
  int cb = 0;
  for (int kt = 0; kt < ntiles; ++kt) {
    if (kt + 1 < ntiles) issue(cb ^ 1);  // prefetch next tile, other buffer
    const float* A = lA[cb];
    const float* Bm = lB[cb];
#pragma unroll
    for (int kc = 0; kc < 16; kc += 4) {
      v2f a0 = *(const v2f*)&A[(wave * 32 + l16) * 20 + kc + hlf * 2];
      v2f a1 = *(const v2f*)&A[(wave * 32 + 16 + l16) * 20 + kc + hlf * 2];
#pragma unroll
      for (int t = 0; t < 8; ++t) {
        v2f b = *(const v2f*)&Bm[(t * 16 + l16) * 20 + kc + hlf * 2];
        acc[0][t] = wmma_f32(a0, b, acc[0][t]);
        acc[1][t] = wmma_f32(a1, b, acc[1][t]);
      }
    }
    wait_async0();
    __syncthreads();
    cb ^= 1;
  }

#pragma unroll
  for (int ms = 0; ms < 2; ++ms) {
#pragma unroll
    for (int t = 0; t < 8; ++t) {
      int col = n0 + t * 16 + l16;
#pragma unroll
      for (int j = 0; j < 8; ++j) {
        int row = m0 + wave * 32 + ms * 16 + j + hlf * 8;
        C[(size_t)row * N + col] = acc[ms][t][j];
      }
    }
  }
}

// ---------------------------------------------------------------------------
// q epilogue: RoPE on last 64 cols of each head's 192, then *SCALE everywhere.
// pos index = b*S + s = bs (matches reference reshape of (B*S, DR) tables).
// ---------------------------------------------------------------------------
__global__ __launch_bounds__(256) void mla_qpost(float* __restrict__ q,
                                                 const float* __restrict__ cosT,
                                                 const float* __restrict__ sinT) {
  unsigned gid = blockIdx.x * 256u + threadIdx.x;
  unsigned i = gid % 160u;  // 128 nope elems + 32 rope pairs
  unsigned bh = gid / 160u;
  unsigned h = bh % 64u;
  unsigned bs = bh / 64u;
  if (bs >= 2048u) return;
  float* row = q + (size_t)bs * 12288 + h * 192;
  if (i < 128u) {
    row[i] *= SCALE_F;
  } else {
    unsigned p = i - 128u;  // 0..31
    const float* cr = cosT + (size_t)bs * 64;
    const float* sr = sinT + (size_t)bs * 64;
    float x0 = row[128 + p];
    float x1 = row[160 + p];
    float o0 = x0 * cr[p] - x1 * sr[p];            // d < 32: rot = -x[d+32]
    float o1 = x1 * cr[p + 32] + x0 * sr[p + 32];  // d >= 32: rot = x[d-32]
    row[128 + p] = o0 * SCALE_F;
    row[160 + p] = o1 * SCALE_F;
  }
}

// k_pe = RoPE(PE)  (unscaled; SCALE folded into q)
__global__ __launch_bounds__(256) void mla_kpe(const float* __restrict__ PE,
                                               const float* __restrict__ cosT,
                                               const float* __restrict__ sinT,
                                               float* __restrict__ kpe) {
  unsigned gid = blockIdx.x * 256u + threadIdx.x;
  unsigned p = gid % 32u;
  unsigned bs = gid / 32u;
  if (bs >= 2048u) return;
  const float* x = PE + (size_t)bs * 64;
  const float* cr = cosT + (size_t)bs * 64;
  const float* sr = sinT + (size_t)bs * 64;
  float x0 = x[p], x1 = x[p + 32];
  kpe[(size_t)bs * 64 + p] = x0 * cr[p] - x1 * sr[p];
  kpe[(size_t)bs * 64 + p + 32] = x1 * cr[p + 32] + x0 * sr[p + 32];
}

// ---------------------------------------------------------------------------
// Causal flash attention, fp32 WMMA. grid = (S/128, B*H), block 256 (8 waves).
// Wave owns 16 query rows; q fragments (16x192) resident in VGPRs.
// K/V tiles (16 keys) double-buffered in LDS via async-to-LDS prefetch.
// Per tile: 48 WMMAs QK^T (2 interleaved chains), online softmax via 16-lane
// xor shuffles, per-wave LDS transpose of P, 32 WMMAs for P*V.
// ---------------------------------------------------------------------------
__global__ __launch_bounds__(256) void mla_attn(const float* __restrict__ q,
                                                const float* __restrict__ kv,
                                                const float* __restrict__ kpe,
                                                float* __restrict__ out) {
  __shared__ float lK[2][16 * 196];   // [key][d] d=0..191 (nope|pe), stride 196
  __shared__ float lV[2][16 * 136];   // [key][dv], stride 136
  __shared__ float lP[8 * 16 * 20];   // per-wave P transpose scratch
  const int tid = threadIdx.x;
  const int wave = tid >> 5;
  const int lane = tid & 31;
  const int hlf = lane >> 4;
  const int l16 = lane & 15;
  const int b = blockIdx.y >> 6;
  const int h = blockIdx.y & 63;
  const int q0 = blockIdx.x * 128;
  const int qrow_w = q0 + wave * 16;

  // Resident q fragments: 48 x v2f = 96 VGPRs (q already scaled+roped)
  v2f qf[48];
  {
    const float* qp =
        q + (size_t)(b * 1024 + qrow_w + l16) * 12288 + h * 192 + hlf * 2;
#pragma unroll
    for (int f = 0; f < 48; ++f) qf[f] = *(const v2f*)(qp + f * 4);
  }

  v8f o[8];
#pragma unroll
  for (int t = 0; t < 8; ++t) o[t] = v8f_zero();
  float m_i[8], l_i[8];
#pragma unroll
  for (int j = 0; j < 8; ++j) { m_i[j] = -3.0e38f; l_i[j] = 0.f; }

  // Async staging: wave w stages keys {2w, 2w+1} each tile.
  // Per key: b128 k_nope (128f), b64 k_pe (64f), b128 v (128f).
  const int keyA = wave * 2;
  const float* gK0 = kv + (size_t)(b * 1024 + keyA) * 16384 + h * 256 + lane * 4;
  const float* gK1 = gK0 + 16384;
  const float* gV0 = gK0 + 128;
  const float* gV1 = gK1 + 128;
  const float* gP0 = kpe + (size_t)(b * 1024 + keyA) * 64 + lane * 2;
  const float* gP1 = gP0 + 64;
  const uint32_t lk0 = lds_off(&lK[0][keyA * 196]) + lane * 16;
  const uint32_t lk1 = lk0 + 196 * 4;
  const uint32_t lpe0 = lds_off(&lK[0][keyA * 196 + 128]) + lane * 8;
  const uint32_t lpe1 = lpe0 + 196 * 4;
  const uint32_t lv0 = lds_off(&lV[0][keyA * 136]) + lane * 16;
  const uint32_t lv1 = lv0 + 136 * 4;
  const uint32_t KBUF = 16 * 196 * 4;
  const uint32_t VBUF = 16 * 136 * 4;

  auto issue_tile = [&](int dstbuf) {
    uint32_t ko = (uint32_t)dstbuf * KBUF;
    uint32_t vo = (uint32_t)dstbuf * VBUF;
    async_ld_b128(lk0 + ko, gK0);
    async_ld_b128(lk1 + ko, gK1);
    async_ld_b64(lpe0 + ko, gP0);
    async_ld_b64(lpe1 + ko, gP1);
    async_ld_b128(lv0 + vo, gV0);
    async_ld_b128(lv1 + vo, gV1);
    gK0 += 16 * 16384; gK1 += 16 * 16384;
    gV0 += 16 * 16384; gV1 += 16 * 16384;
    gP0 += 16 * 64; gP1 += 16 * 64;
  };

  float* pw = &lP[wave * 320];
  const int kt_end = (q0 >> 4) + 8;  // keys < q0+128

  issue_tile(0);
  wait_async0();
  __syncthreads();

  int cb = 0;
  for (int kt = 0; kt < kt_end; ++kt) {
    const int key0 = kt * 16;
    if (kt + 1 < kt_end) issue_tile(cb ^ 1);
    const float* Kt = lK[cb];
    const float* Vt = lV[cb];

    if (key0 <= qrow_w + 15) {  // wave-uniform: skip fully-masked tiles
      // S = q * K^T  (d = 192 -> 48 fp32 WMMAs, two interleaved chains)
      v8f s0 = v8f_zero(), s1 = v8f_zero();
#pragma unroll
      for (int f = 0; f < 48; f += 2) {
        v2f bf0 = *(const v2f*)&Kt[l16 * 196 + f * 4 + hlf * 2];
        v2f bf1 = *(const v2f*)&Kt[l16 * 196 + (f + 1) * 4 + hlf * 2];
        s0 = wmma_f32(qf[f], bf0, s0);
        s1 = wmma_f32(qf[f + 1], bf1, s1);
      }
      v8f s;
#pragma unroll
      for (int j = 0; j < 8; ++j) s[j] = s0[j] + s1[j];
      // causal mask
      const int keyc = key0 + l16;
#pragma unroll
      for (int j = 0; j < 8; ++j) {
        int qr = qrow_w + j + hlf * 8;
        if (keyc > qr) s[j] = -1.0e30f;
      }
      // online softmax (row reduce across 16-lane halves)
#pragma unroll
      for (int j = 0; j < 8; ++j) {
        float v = s[j];
        v = fmaxf(v, __shfl_xor(v, 1, 32));
        v = fmaxf(v, __shfl_xor(v, 2, 32));
        v = fmaxf(v, __shfl_xor(v, 4, 32));
        v = fmaxf(v, __shfl_xor(v, 8, 32));
        float mnew = fmaxf(m_i[j], v);
        float corr = __expf(m_i[j] - mnew);
        m_i[j] = mnew;
        float p = __expf(s[j] - mnew);
        s[j] = p;
        float rs = p;
        rs += __shfl_xor(rs, 1, 32);
        rs += __shfl_xor(rs, 2, 32);
        rs += __shfl_xor(rs, 4, 32);
        rs += __shfl_xor(rs, 8, 32);
        l_i[j] = l_i[j] * corr + rs;
#pragma unroll
        for (int t = 0; t < 8; ++t) o[t][j] *= corr;
      }
      // transpose P (C-layout -> A-layout) via wave-private LDS (in-order DS)
#pragma unroll
      for (int j = 0; j < 8; ++j) pw[(j + hlf * 8) * 20 + l16] = s[j];
      v2f a4[4];
#pragma unroll
      for (int kc4 = 0; kc4 < 4; ++kc4)
        a4[kc4] = *(const v2f*)&pw[l16 * 20 + kc4 * 4 + hlf * 2];
      // O += P * V  (32 fp32 WMMAs; V row-major, B-frag from two b32 loads)
#pragma unroll
      for (int t = 0; t < 8; ++t) {
#pragma unroll
        for (int kc4 = 0; kc4 < 4; ++kc4) {
          int kr = kc4 * 4 + hlf * 2;
          v2f bf;
          bf.x = Vt[kr * 136 + t * 16 + l16];
          bf.y = Vt[(kr + 1) * 136 + t * 16 + l16];
          o[t] = wmma_f32(a4[kc4], bf, o[t]);
        }
      }
    }
    wait_async0();
    __syncthreads();
    cb ^= 1;
  }

  // epilogue: O / l  -> out[b, q, h, dv]
#pragma unroll
  for (int j = 0; j < 8; ++j) {
    int qr = qrow_w + j + hlf * 8;
    float inv = 1.0f / l_i[j];
    size_t base = (size_t)(b * 1024 + qr) * 8192 + h * 128;
#pragma unroll
    for (int t = 0; t < 8; ++t) out[base + t * 16 + l16] = o[t][j] * inv;
  }
}

extern "C" void kernel_launch(void* const* d_in, const int* in_sizes, int n_in,
                              void* d_out, int out_size, void* d_ws,
                              size_t ws_size, hipStream_t stream) {
  const float* Q = (const float*)d_in[0];     // (2, 1024, 1536)
  const float* KV = (const float*)d_in[1];    // (2, 1024, 512)
  const float* PE = (const float*)d_in[2];    // (2, 1024, 64)
  const float* WUQ = (const float*)d_in[3];   // (12288, 1536)
  const float* WUKV = (const float*)d_in[4];  // (16384, 512)
  const float* cosT = (const float*)d_in[5];  // (2048, 64)
  const float* sinT = (const float*)d_in[6];  // (2048, 64)
  float* out = (float*)d_out;                 // (2, 1024, 64, 128)

  float* qraw = (float*)d_ws;                  // 2048 x 12288
  float* kvraw = qraw + (size_t)2048 * 12288;  // 2048 x 16384
  float* kpe = kvraw + (size_t)2048 * 16384;   // 2048 x 64

  dim3 blk(256);
  mla_gemm_nt<<<dim3(96, 8), blk, 0, stream>>>(Q, WUQ, qraw, 1536, 12288);
  mla_gemm_nt<<<dim3(128, 8), blk, 0, stream>>>(KV, WUKV, kvraw, 512, 16384);
  mla_kpe<<<dim3(256), blk, 0, stream>>>(PE, cosT, sinT, kpe);
  mla_qpost<<<dim3(81920), blk, 0, stream>>>(qraw, cosT, sinT);
  mla_attn<<<dim3(8, 128), blk, 0, stream>>>(qraw, kvraw, kpe, out);
}